// OTLoss_65017214927533
// MI455X (gfx1250) — compile-verified
//
#include <hip/hip_runtime.h>
#include <cmath>
#include <vector>

typedef __attribute__((ext_vector_type(2))) float v2f;
typedef __attribute__((ext_vector_type(4))) float v4f;
typedef __attribute__((ext_vector_type(8))) float v8f;

#define NPTS 4096
#define DIM  64
#define SPLITS 4
#define COLS_PER_SPLIT (NPTS / SPLITS)
#define LOG2E 1.44269504088896340736f
#define LN2   0.69314718055994530942f

// workspace float offsets
#define OFF_X2  0
#define OFF_Y2  (NPTS)
#define OFF_POT (2*NPTS)    // f_ba, g_ab, f_aa, g_bb
#define OFF_FIN (6*NPTS)    // f_n, g_n, faa_n, gbb_n
#define OFF_PQ  (10*NPTS)   // per-job fused pot[j] - 0.5*csq[j]
#define OFF_PM  (14*NPTS)   // partial max  [job][split][row]  (4*SPLITS*N)
#define OFF_PS  (OFF_PM + 4*SPLITS*NPTS)  // partial sum (exp2 domain)
// total: OFF_PS + 4*SPLITS*NPTS = 46*NPTS floats  (~736 KB)

// ---------------------------------------------------------------------------
// squared norms of every point + zero the potential arrays
__global__ void prep_kernel(const float* __restrict__ X, const float* __restrict__ Y,
                            float* __restrict__ ws) {
    int t = blockIdx.x * blockDim.x + threadIdx.x;
    if (t < 4 * NPTS) ws[OFF_POT + t] = 0.0f;
    if (t < 2 * NPTS) {
        const float* src = (t < NPTS) ? (X + (size_t)t * DIM) : (Y + (size_t)(t - NPTS) * DIM);
        float s = 0.0f;
#pragma unroll
        for (int k = 0; k < DIM; k += 4) {
            v4f v = *(const v4f*)(src + k);
            s += v.x * v.x + v.y * v.y + v.z * v.z + v.w * v.w;
        }
        ws[OFF_X2 + t] = s;   // X2 then Y2, contiguous
    }
}

// ---------------------------------------------------------------------------
// one-time init of pq_job[j] = pot_job[j] - 0.5*csq_job[j]  (pot == 0 here)
__global__ void fuse_pq_kernel(float* __restrict__ ws) {
    int t = blockIdx.x * blockDim.x + threadIdx.x;
    if (t < 4 * NPTS) {
        int job = t >> 12;
        int c   = t & (NPTS - 1);
        const float* csq = (job == 0 || job == 3) ? (ws + OFF_Y2) : (ws + OFF_X2);
        ws[OFF_PQ + t] = ws[OFF_POT + t] - 0.5f * csq[c];
    }
}

// ---------------------------------------------------------------------------
// Fused cost-recompute + partial row-softmin over a 1024-column slab.
// Produces per-(job,split,row) online-LSE partials (M in exp2 domain, S).
// C_ij = 0.5*rsq[i] + 0.5*csq[j] - dot(A_i,B_j) via V_WMMA_F32_16X16X4_F32.
// blockIdx.y = job (0:xy 1:yx 2:xx 3:yy), blockIdx.z = column split.
__global__ __launch_bounds__(256)
void softmin_part(const float* __restrict__ X, const float* __restrict__ Y,
                  float* __restrict__ ws, float eps) {
    const int lane   = threadIdx.x & 31;
    const int wave   = threadIdx.x >> 5;
    const int rowBlk = blockIdx.x * 8 + wave;   // 0..255
    const int job    = blockIdx.y;              // 0..3
    const int split  = blockIdx.z;              // 0..SPLITS-1

    const float *A, *B, *rsq;
    switch (job) {
      case 0:  A = X; B = Y; rsq = ws + OFF_X2; break;
      case 1:  A = Y; B = X; rsq = ws + OFF_Y2; break;
      case 2:  A = X; B = X; rsq = ws + OFF_X2; break;
      default: A = Y; B = Y; rsq = ws + OFF_Y2; break;
    }
    const float* __restrict__ pq = ws + OFF_PQ + job * NPTS;

    const float inv_eps = 1.0f / eps;
    const float h_const = -logf((float)NPTS);     // a_log == b_log
    const float scale   = inv_eps * LOG2E;        // exp2 domain

    const int R    = rowBlk * 16;
    const int half = lane >> 4;
    const int l16  = lane & 15;
    const int c0   = split * COLS_PER_SPLIT;
    const int cEnd = c0 + COLS_PER_SPLIT;

    // Preload the 16x64 A tile (A-layout: lanes<16 rows R+l K=k0,k0+1; lanes>=16 K=k0+2,k0+3)
    v2f a[16];
    {
        const float* arow = A + (size_t)(R + l16) * DIM + 2 * half;
#pragma unroll
        for (int ks = 0; ks < 16; ++ks)
            a[ks] = *(const v2f*)(arow + ks * 4);
    }

    // u[r] = -0.5*rsq[row]*scale for row = R + r + 8*half
    float myrs = rsq[R + l16];
    float u[8];
#pragma unroll
    for (int r = 0; r < 8; ++r) {
        float rv = __shfl(myrs, r + 8 * half, 32);
        u[r] = -0.5f * rv * scale;
    }

    float m[8], s[8];
#pragma unroll
    for (int r = 0; r < 8; ++r) { m[r] = -3.0e38f; s[r] = 0.0f; }

    // one column tile: 16 chained WMMAs + single-exp online-LSE update
    auto compute = [&](const v2f* bt, float pcv) {
        float tcol = fmaf(pcv, inv_eps, h_const) * LOG2E;
        v8f acc = {};
#pragma unroll
        for (int ks = 0; ks < 16; ++ks)
            acc = __builtin_amdgcn_wmma_f32_16x16x4_f32(
                false, a[ks], false, bt[ks], (short)0, acc, false, false);
#pragma unroll
        for (int r = 0; r < 8; ++r) {
            float v  = fmaf(acc[r], scale, tcol + u[r]);
            float mn = fmaxf(m[r], v);
            float nd = fminf(m[r], v) - mn;               // -(|m-v|) <= 0
            float e  = __builtin_amdgcn_exp2f(nd);        // raw v_exp_f32
            bool  nm = v > m[r];
            float c1 = nm ? e : 1.0f;                     // rescale old sum
            float c2 = nm ? 1.0f : e;                     // new term
            s[r] = fmaf(s[r], c1, c2);
            m[r] = mn;
        }
    };

    // -------- software-pipelined column loop (ping-pong B buffers) ---------
    v2f b0[16], b1[16];
    float pc0, pc1;
    const float* bptr = B + (size_t)(c0 + l16) * DIM + 2 * half;

    pc0 = pq[c0 + l16];
#pragma unroll
    for (int ks = 0; ks < 16; ++ks) b0[ks] = *(const v2f*)(bptr + ks * 4);
    bptr += 16 * DIM;

    for (int Cb = c0; Cb < cEnd; Cb += 32) {
        pc1 = pq[Cb + 16 + l16];
#pragma unroll
        for (int ks = 0; ks < 16; ++ks) b1[ks] = *(const v2f*)(bptr + ks * 4);
        bptr += 16 * DIM;

        compute(b0, pc0);

        if (Cb + 32 < cEnd) {
            pc0 = pq[Cb + 32 + l16];
#pragma unroll
            for (int ks = 0; ks < 16; ++ks) b0[ks] = *(const v2f*)(bptr + ks * 4);
            bptr += 16 * DIM;
        }

        compute(b1, pc1);
    }

    // Reduce (m,s) across the 16 lanes of each half; store per-row partials.
    float* PM = ws + OFF_PM + (size_t)(job * SPLITS + split) * NPTS;
    float* PS = ws + OFF_PS + (size_t)(job * SPLITS + split) * NPTS;
#pragma unroll
    for (int r = 0; r < 8; ++r) {
        float mr = m[r], sr = s[r];
#pragma unroll
        for (int d = 8; d >= 1; d >>= 1) {
            float mo = __shfl_xor(mr, d, 32);
            float so = __shfl_xor(sr, d, 32);
            float mn = fmaxf(mr, mo);
            sr = sr * __builtin_amdgcn_exp2f(mr - mn)
               + so * __builtin_amdgcn_exp2f(mo - mn);
            mr = mn;
        }
        if (l16 == 0) {
            int row = R + r + 8 * half;
            PM[row] = mr;
            PS[row] = sr;
        }
    }
}

// ---------------------------------------------------------------------------
// Merge SPLITS partials -> softmin value; then either
//   finalFlag=0: pot = wOld*pot + (1-wOld)*res  and refresh pq
//   finalFlag=1: write res into FIN
__global__ void finish_kernel(float* __restrict__ ws, float eps, float wOld, int finalFlag) {
    int t = blockIdx.x * blockDim.x + threadIdx.x;
    if (t >= 4 * NPTS) return;
    int job = t >> 12;
    int row = t & (NPTS - 1);

    const float* PM = ws + OFF_PM + (size_t)job * SPLITS * NPTS;
    const float* PS = ws + OFF_PS + (size_t)job * SPLITS * NPTS;

    float M = -3.0e38f, S = 0.0f;
#pragma unroll
    for (int k = 0; k < SPLITS; ++k) {
        float mk = PM[k * NPTS + row];
        float sk = PS[k * NPTS + row];
        float mn = fmaxf(M, mk);
        S = S * __builtin_amdgcn_exp2f(M - mn)
          + sk * __builtin_amdgcn_exp2f(mk - mn);
        M = mn;
    }
    float lse = (M + __builtin_log2f(S)) * LN2;   // natural-log LSE
    float res = -eps * lse;

    if (finalFlag) {
        ws[OFF_FIN + t] = res;
    } else {
        float p = wOld * ws[OFF_POT + t] + (1.0f - wOld) * res;
        ws[OFF_POT + t] = p;
        const float* csq = (job == 0 || job == 3) ? (ws + OFF_Y2) : (ws + OFF_X2);
        ws[OFF_PQ + t] = p - 0.5f * csq[row];
    }
}

// ---------------------------------------------------------------------------
// result = (1/N) * ( sum f_n + sum g_n - sum faa_n - sum gbb_n )
__global__ void reduce_kernel(const float* __restrict__ ws, float* __restrict__ outp) {
    __shared__ float sm[256];
    int t = threadIdx.x;
    float acc = 0.0f;
    for (int i = t; i < 4 * NPTS; i += 256) {
        float v = ws[OFF_FIN + i];
        acc += (i < 2 * NPTS) ? v : -v;
    }
    sm[t] = acc;
    __syncthreads();
    for (int d = 128; d > 0; d >>= 1) {
        if (t < d) sm[t] += sm[t + d];
        __syncthreads();
    }
    if (t == 0) outp[0] = sm[0] * (1.0f / (float)NPTS);
}

// ---------------------------------------------------------------------------
extern "C" void kernel_launch(void* const* d_in, const int* in_sizes, int n_in,
                              void* d_out, int out_size, void* d_ws, size_t ws_size,
                              hipStream_t stream) {
    const float* X = (const float*)d_in[0];
    const float* Y = (const float*)d_in[1];
    float* out = (float*)d_out;
    float* ws  = (float*)d_ws;

    // geomloss epsilon_schedule(p=2, diameter=20, blur=0.01, scaling=0.7)
    const double P = 2.0, diam = 20.0, blur = 0.01, scal = 0.7;
    std::vector<float> eps_list;
    eps_list.push_back((float)std::pow(diam, P));
    for (double e = P * std::log(diam); e > P * std::log(blur); e += P * std::log(scal))
        eps_list.push_back((float)std::exp(e));
    eps_list.push_back((float)std::pow(blur, P));

    const dim3 blk(256);
    const dim3 gSoft(32, 4, SPLITS);         // 256 row-blocks/8 waves, 4 jobs, 4 col slabs
    const int  nVec = (4 * NPTS + 255) / 256;

    prep_kernel<<<nVec, blk, 0, stream>>>(X, Y, ws);
    fuse_pq_kernel<<<nVec, blk, 0, stream>>>(ws);

    // init: potentials are zero -> softmin with plain a_log/b_log; assign (wOld=0)
    softmin_part<<<gSoft, blk, 0, stream>>>(X, Y, ws, eps_list[0]);
    finish_kernel<<<nVec, blk, 0, stream>>>(ws, eps_list[0], 0.0f, 0);

    // symmetric Sinkhorn updates with averaging over the full eps schedule
    for (size_t it = 0; it < eps_list.size(); ++it) {
        softmin_part<<<gSoft, blk, 0, stream>>>(X, Y, ws, eps_list[it]);
        finish_kernel<<<nVec, blk, 0, stream>>>(ws, eps_list[it], 0.5f, 0);
    }

    // final extrapolation at eps = blur^p (detached potentials; same forward math)
    softmin_part<<<gSoft, blk, 0, stream>>>(X, Y, ws, eps_list.back());
    finish_kernel<<<nVec, blk, 0, stream>>>(ws, eps_list.back(), 0.0f, 1);

    reduce_kernel<<<1, blk, 0, stream>>>(ws, out);
}